// MyNetwork_30167850287769
// MI455X (gfx1250) — compile-verified
//
#include <hip/hip_runtime.h>

typedef __attribute__((ext_vector_type(2))) float v2f;
typedef __attribute__((ext_vector_type(8))) float v8f;

#define N_GRAPHS 64
#define OUTC2 16

// ---------------------------------------------------------------------------
// deg[i] = 1 (self loop); also zero the pooled output (harness poisons it)
__global__ void k_init(float* __restrict__ deg, float* __restrict__ out, int nN) {
  int i = blockIdx.x * blockDim.x + threadIdx.x;
  if (i < nN) deg[i] = 1.0f;
  if (i < N_GRAPHS * OUTC2) out[i] = 0.0f;
}

// deg[dst[e]] += 1
__global__ void k_deg(const int* __restrict__ dst, float* __restrict__ deg, int nE) {
  int e = blockIdx.x * blockDim.x + threadIdx.x;
  if (e < nE) atomicAdd(&deg[dst[e]], 1.0f);
}

// deg -> rsqrt(deg) in place (deg >= 1 always, so no zero guard needed)
__global__ void k_dinv(float* __restrict__ deg, int nN) {
  int i = blockIdx.x * blockDim.x + threadIdx.x;
  if (i < nN) deg[i] = rsqrtf(deg[i]);
}

// norm[e] = dinv[src]*dinv[dst]  (reused by both layers)
__global__ void k_norm(const int* __restrict__ src, const int* __restrict__ dst,
                       const float* __restrict__ dinv, float* __restrict__ norm, int nE) {
  int e = blockIdx.x * blockDim.x + threadIdx.x;
  if (e < nE) norm[e] = dinv[src[e]] * dinv[dst[e]];
}

// ---------------------------------------------------------------------------
// Dense projection H = act(X + bias?) @ W using V_WMMA_F32_16X16X4_F32.
// One wave per 16-row tile. A: 16x4 f32 (2 VGPRs), B: 4x16 (2 VGPRs),
// C/D: 16x16 f32 (8 VGPRs). Layouts per CDNA5 ISA §7.12.2.
template <int IN_C, int OUT_C, bool RELU_BIAS>
__global__ void k_gemm_wmma(const float* __restrict__ X, const float* __restrict__ W,
                            const float* __restrict__ bias, float* __restrict__ H,
                            int nTiles) {
  const int wv   = threadIdx.x >> 5;
  const int lane = threadIdx.x & 31;
  const int tile = blockIdx.x * (blockDim.x >> 5) + wv;
  if (tile >= nTiles) return;            // wave-uniform: EXEC all-1s inside wave
  const int base = tile * 16;
  const int r    = lane & 15;            // M index (A) / N index (B,D)
  const int kg   = lane >> 4;            // lane-half selects K pair / row half

  v8f acc[OUT_C / 16];
#pragma unroll
  for (int nt = 0; nt < OUT_C / 16; ++nt) { v8f z = {}; acc[nt] = z; }

#pragma unroll
  for (int k0 = 0; k0 < IN_C; k0 += 4) {
    const int ka = k0 + 2 * kg;          // VGPR0 holds K=ka, VGPR1 holds K=ka+1
    v2f a;
    if constexpr (RELU_BIAS) {
      float ax = X[(base + r) * IN_C + ka]     + bias[ka];
      float ay = X[(base + r) * IN_C + ka + 1] + bias[ka + 1];
      a.x = ax > 0.0f ? ax : 0.0f;
      a.y = ay > 0.0f ? ay : 0.0f;
    } else {
      a = *(const v2f*)(X + (base + r) * IN_C + ka);   // 8B-aligned (ka even)
    }
#pragma unroll
    for (int nt = 0; nt < OUT_C / 16; ++nt) {
      v2f b;
      b.x = W[ ka      * OUT_C + nt * 16 + r];
      b.y = W[(ka + 1) * OUT_C + nt * 16 + r];
      acc[nt] = __builtin_amdgcn_wmma_f32_16x16x4_f32(
          false, a, false, b, (short)0, acc[nt], false, false);
    }
  }

  // D layout: VGPR v holds row M = v + 8*kg, col N = nt*16 + r
#pragma unroll
  for (int nt = 0; nt < OUT_C / 16; ++nt)
#pragma unroll
    for (int v = 0; v < 8; ++v)
      H[(base + v + 8 * kg) * OUT_C + nt * 16 + r] = acc[nt][v];
}

// ---------------------------------------------------------------------------
// agg[i,:] = h[i,:] * dinv[i]^2   (the self-loop message; also zero-inits agg)
template <int C>
__global__ void k_selfinit(const float* __restrict__ h, const float* __restrict__ dinv,
                           float* __restrict__ agg, int nN) {
  int t = blockIdx.x * blockDim.x + threadIdx.x;
  int i = t / (C / 4);
  if (i >= nN) return;
  int c = (t % (C / 4)) * 4;
  float s = dinv[i]; s *= s;
  float4 v = *(const float4*)(h + i * C + c);
  float4 o; o.x = v.x * s; o.y = v.y * s; o.z = v.z * s; o.w = v.w * s;
  *(float4*)(agg + i * C + c) = o;
}

// agg[dst[e],:] += h[src[e],:] * norm[e]   (C/4 lanes per edge, float4 gather)
template <int C>
__global__ void k_scatter(const int* __restrict__ src, const int* __restrict__ dst,
                          const float* __restrict__ norm, const float* __restrict__ h,
                          float* __restrict__ agg, int nE) {
  const int LPE = C / 4;
  int t = blockIdx.x * blockDim.x + threadIdx.x;
  int e = t / LPE;
  if (e >= nE) return;
  int c = (t % LPE) * 4;
  int s = src[e], d = dst[e];
  float w = norm[e];
  float4 v = *(const float4*)(h + s * C + c);
  float* p = agg + d * C + c;
  atomicAdd(p + 0, v.x * w);
  atomicAdd(p + 1, v.y * w);
  atomicAdd(p + 2, v.z * w);
  atomicAdd(p + 3, v.w * w);
}

// ---------------------------------------------------------------------------
// out[batch[i],:] += agg2[i,:] + b2  — LDS-hierarchical (ds_add_f32) then
// flush only non-zero graph slots (batch is sorted -> few graphs per block).
__global__ void k_pool(const float* __restrict__ agg2, const float* __restrict__ b2,
                       const int* __restrict__ batch, float* __restrict__ out, int nN) {
  __shared__ float acc[N_GRAPHS * OUTC2];
  for (int j = threadIdx.x; j < N_GRAPHS * OUTC2; j += blockDim.x) acc[j] = 0.0f;
  __syncthreads();
  int i = blockIdx.x * blockDim.x + threadIdx.x;
  if (i < nN) {
    int g = batch[i];
    const float* row = agg2 + i * OUTC2;
#pragma unroll
    for (int c = 0; c < OUTC2; ++c) atomicAdd(&acc[g * OUTC2 + c], row[c] + b2[c]);
  }
  __syncthreads();
  for (int j = threadIdx.x; j < N_GRAPHS * OUTC2; j += blockDim.x) {
    float v = acc[j];
    if (v != 0.0f) atomicAdd(&out[j], v);
  }
}

// ---------------------------------------------------------------------------
extern "C" void kernel_launch(void* const* d_in, const int* in_sizes, int n_in,
                              void* d_out, int out_size, void* d_ws, size_t ws_size,
                              hipStream_t stream) {
  const float* x     = (const float*)d_in[0];
  const int*   ei    = (const int*)d_in[1];   // [2,E] int32 (JAX demotes int64)
  const int*   batch = (const int*)d_in[2];
  const float* W1    = (const float*)d_in[3];
  const float* b1    = (const float*)d_in[4];
  const float* W2    = (const float*)d_in[5];
  const float* b2    = (const float*)d_in[6];
  float* out = (float*)d_out;

  const int nN = in_sizes[2];        // 100000 (batch has one entry per node)
  const int nE = in_sizes[1] / 2;    // 1600000
  const int* src = ei;
  const int* dst = ei + nE;

  // Workspace carve-up (floats, 256B-aligned regions); ~45 MB total.
  float* ws = (float*)d_ws;
  size_t off = 0;
  auto take = [&](size_t nf) { float* p = ws + off; off += (nf + 63) & ~(size_t)63; return p; };
  float* dinv = take((size_t)nN);        // deg, then rsqrt in place
  float* norm = take((size_t)nE);
  float* h1   = take((size_t)nN * 32);
  float* agg1 = take((size_t)nN * 32);
  float* h2   = take((size_t)nN * 16);
  float* agg2 = take((size_t)nN * 16);
  (void)ws_size; (void)n_in; (void)out_size;

  const int B  = 256;
  const int gN = (nN + B - 1) / B;
  const int gE = (nE + B - 1) / B;
  const int nTiles = (nN + 15) / 16;
  const int gT = (nTiles + 3) / 4;       // 4 waves (tiles) per 128-thread block

  k_init<<<gN, B, 0, stream>>>(dinv, out, nN);
  k_deg <<<gE, B, 0, stream>>>(dst, dinv, nE);
  k_dinv<<<gN, B, 0, stream>>>(dinv, nN);
  k_norm<<<gE, B, 0, stream>>>(src, dst, dinv, norm, nE);

  // Layer 1: h1 = x @ W1 ; agg1 = scatter(norm * h1) incl. self loops
  k_gemm_wmma<32, 32, false><<<gT, 128, 0, stream>>>(x, W1, nullptr, h1, nTiles);
  k_selfinit<32><<<(nN * 8 + B - 1) / B, B, 0, stream>>>(h1, dinv, agg1, nN);
  k_scatter <32><<<(nE * 8 + B - 1) / B, B, 0, stream>>>(src, dst, norm, h1, agg1, nE);

  // Layer 2: h2 = relu(agg1 + b1) @ W2 ; agg2 = scatter(norm * h2)
  k_gemm_wmma<32, 16, true><<<gT, 128, 0, stream>>>(agg1, W2, b1, h2, nTiles);
  k_selfinit<16><<<(nN * 4 + B - 1) / B, B, 0, stream>>>(h2, dinv, agg2, nN);
  k_scatter <16><<<(nE * 4 + B - 1) / B, B, 0, stream>>>(src, dst, norm, h2, agg2, nE);

  // Pool (+b2 folded in)
  k_pool<<<gN, B, 0, stream>>>(agg2, b2, batch, out, nN);
}